// EPGModule_76115410420435
// MI455X (gfx1250) — compile-verified
//
#include <hip/hip_runtime.h>
#include <math.h>

// Problem constants (match reference)
#define NECH 32      // echoes / EPG orders
#define NPC  40      // compartments
#define PC   48      // compartments padded to multiple of 4 (and of 16*3) for WMMA K-chunks
#define BVOX 256     // voxels
#define EPG_SIZE 97  // 3*NECH + 1

typedef __attribute__((ext_vector_type(2))) float v2f;
typedef __attribute__((ext_vector_type(8))) float v8f;

// -----------------------------------------------------------------------------
// Kernel 1: structured EPG recursion. One lane per (voxel, compartment) pair.
// State x[97] lives entirely in VGPRs; one step is
//   z = T * (d .* (S x));   x = d .* (S z)
// with S a sparse shift (<=1 nonzero/row), T block-diagonal 3x3 rotations,
// d = [e2, (e2,e2,e1)*32] the relaxation diagonal. ~700 VALU ops / step.
// -----------------------------------------------------------------------------
__global__ __launch_bounds__(256) void epg_recursion_kernel(
    const float* __restrict__ refoc,      // (BVOX)
    const float* __restrict__ t2s,        // (BVOX, NPC)
    const float* __restrict__ weights,    // (BVOX, NPC)
    float* __restrict__ H_ws,             // (BVOX, NECH, PC)
    float* __restrict__ W_ws)             // (BVOX, PC)
{
    const int t = blockIdx.x * blockDim.x + threadIdx.x;
    if (t >= BVOX * PC) return;
    const int b = t / PC;
    const int c = t % PC;

    // padded weights for the WMMA reduction (zeros in the K-pad region)
    W_ws[b * PC + c] = (c < NPC) ? weights[b * NPC + c] : 0.0f;

    float* __restrict__ Hb = H_ws + b * NECH * PC;
    if (c >= NPC) {
        #pragma unroll
        for (int e = 0; e < NECH; ++e) Hb[e * PC + c] = 0.0f;
        return;
    }

    // Per-voxel rotation terms. alpha_exc = alpha/2, so x0 reuses sin/cos(alpha/2).
    const float alpha = refoc[b] * 0.017453292519943295f;
    const float sh = __sinf(0.5f * alpha);
    const float ch = __cosf(0.5f * alpha);
    const float c2 = ch * ch;           // cos^2(a/2)
    const float s2 = sh * sh;           // sin^2(a/2)
    const float ss = 2.0f * sh * ch;    // sin(a)
    const float cc = c2 - s2;           // cos(a)

    // Relaxation: tau = TAU/2 = 5 ms. e1 = exp(-5/1000) is a compile-time constant.
    const float e2 = __expf(-5.0f / t2s[b * NPC + c]);
    const float e1 = 0.9950124791926823f; // exp(-0.005)

    float x[EPG_SIZE];
    #pragma unroll
    for (int i = 0; i < EPG_SIZE; ++i) x[i] = 0.0f;
    x[0] = sh;   // sin(alpha_exc)
    x[2] = ch;   // cos(alpha_exc)

    #pragma unroll 1
    for (int e = 0; e < NECH; ++e) {
        float z[EPG_SIZE];
        // ---- z = T * (d .* (S x)) ----
        z[0] = e2 * x[2];                              // row 0: S[0,2], T[0,0]=1
        #pragma unroll
        for (int k = 0; k < NECH; ++k) {
            const int o = k + 1;                       // EPG order, rows 3o-2..3o
            const float ya = e2 * ((o == 1) ? x[0] : x[3 * o - 5]);
            const float yb = (o <= 31)
                ? (e2 * ((o == 1) ? x[5] : x[3 * o + 2]))
                : 0.0f;                                // row 95 of S is empty
            const float yc = e1 * x[3 * o];
            z[3 * o - 2] =  c2 * ya + s2 * yb + ss * yc;
            z[3 * o - 1] =  s2 * ya + c2 * yb - ss * yc;
            z[3 * o]     = -0.5f * ss * ya + 0.5f * ss * yb + cc * yc;
        }
        // ---- x = d .* (S z) ----
        x[0] = e2 * z[2];
        #pragma unroll
        for (int k = 0; k < NECH; ++k) {
            const int o = k + 1;
            x[3 * o - 2] = e2 * ((o == 1) ? z[0] : z[3 * o - 5]);
            x[3 * o - 1] = (o <= 31)
                ? (e2 * ((o == 1) ? z[5] : z[3 * o + 2]))
                : 0.0f;
            x[3 * o]     = e1 * z[3 * o];
        }
        Hb[e * PC + c] = x[0];   // echo amplitude after this step
    }
}

// -----------------------------------------------------------------------------
// Kernel 2: per-voxel weighted contraction sig[e] = sum_c H[e,c]*w[c] via
// V_WMMA_F32_16X16X4_F32, then normalize by echo 0. One wave32 per voxel.
//   A (16x4)  : 16 echoes x 4-compartment K-chunk of H  (ISA 7.12.2 layout:
//               lanes 0-15 hold K={0,1}, lanes 16-31 hold K={2,3}, M = lane&15)
//   B (4x16)  : weights in column N=0, zeros elsewhere (mirrored K layout)
//   D (16x16) : column N=0 = sig; lives in lane 0 (M=r) and lane 16 (M=8+r)
// -----------------------------------------------------------------------------
__global__ __launch_bounds__(128) void epg_reduce_wmma_kernel(
    const float* __restrict__ H_ws,   // (BVOX, NECH, PC)
    const float* __restrict__ W_ws,   // (BVOX, PC)
    float* __restrict__ out)          // (BVOX, NECH)
{
    const int wave = threadIdx.x >> 5;
    const int lane = threadIdx.x & 31;
    const int b    = blockIdx.x * 4 + wave;

    const float* __restrict__ Hb = H_ws + b * NECH * PC;
    const float* __restrict__ Wb = W_ws + b * PC;

    const int   m     = lane & 15;                // row (M) / col (N) index
    const int   kk0   = (lane < 16) ? 0 : 2;      // K phase for this lane half
    const float wmask = (m == 0) ? 1.0f : 0.0f;   // only column N=0 of B is live

    __shared__ float sig[4][NECH];

    #pragma unroll
    for (int h = 0; h < 2; ++h) {                 // echo halves: rows 0-15, 16-31
        v8f acc = {};
        #pragma unroll
        for (int k = 0; k < PC / 4; ++k) {        // 12 K-chunks of 4 compartments
            const float* row = Hb + (16 * h + m) * PC + 4 * k + kk0;
            v2f a, bb;
            a.x  = row[0];
            a.y  = row[1];
            bb.x = wmask * Wb[4 * k + kk0];
            bb.y = wmask * Wb[4 * k + kk0 + 1];
            acc = __builtin_amdgcn_wmma_f32_16x16x4_f32(
                /*neg_a=*/false, a, /*neg_b=*/false, bb,
                /*c_mod=*/(short)0, acc, /*reuse_a=*/false, /*reuse_b=*/false);
        }
        if (m == 0) {                              // lanes 0 and 16 hold column N=0
            const int base = 16 * h + ((lane == 16) ? 8 : 0);
            #pragma unroll
            for (int r = 0; r < 8; ++r) sig[wave][base + r] = acc[r];
        }
    }
    __syncthreads();
    const float s0 = sig[wave][0];
    if (lane < NECH) out[b * NECH + lane] = sig[wave][lane] / s0;
}

// -----------------------------------------------------------------------------
extern "C" void kernel_launch(void* const* d_in, const int* in_sizes, int n_in,
                              void* d_out, int out_size, void* d_ws, size_t ws_size,
                              hipStream_t stream) {
    const float* refoc   = (const float*)d_in[0];   // (256,)
    const float* t2s     = (const float*)d_in[1];   // (256,40)
    const float* weights = (const float*)d_in[2];   // (256,40)

    float* H_ws = (float*)d_ws;                     // 256*32*48 floats
    float* W_ws = H_ws + BVOX * NECH * PC;          // 256*48 floats (total ~1.55 MB)

    epg_recursion_kernel<<<(BVOX * PC + 255) / 256, 256, 0, stream>>>(
        refoc, t2s, weights, H_ws, W_ws);
    epg_reduce_wmma_kernel<<<BVOX / 4, 128, 0, stream>>>(
        H_ws, W_ws, (float*)d_out);
}